// Least_square_loss_75290776699470
// MI455X (gfx1250) — compile-verified
//
#include <hip/hip_runtime.h>
#include <math.h>

#define NPTS 2048
#define BLOCK 256
#define VEC4_PER_THREAD 4        // each v4f = 2 points, 4*256*2 = 2048

typedef __attribute__((ext_vector_type(4))) float v4f;
typedef __attribute__((ext_vector_type(2))) float v2f;

// One block per batch. Streams actual points (phase 1: power moments), solves
// the SPD 4x4 system in f64 (Cholesky) on lane 0, then evaluates the loss on
// target points staged in LDS via the CDNA5 async global->LDS path, which is
// issued at kernel entry and overlaps all of phase 1.
__global__ __launch_bounds__(BLOCK) void lsq_loss_kernel(
    const float* __restrict__ tgt,   // (B, P, 2) float32: (x, y)
    const float* __restrict__ act,   // (B, P, 2) float32: (x, y)
    float*       __restrict__ out)   // (B, P)    float32
{
    __shared__ v4f   s_tgt[NPTS / 2];    // 16 KB: target tile (x0,y0,x1,y1)
    __shared__ float s_acc[10];          // m1..m6, X0..X3
    __shared__ float s_beta[4];

    const int b   = blockIdx.x;
    const int tid = threadIdx.x;

    const v4f* gt = (const v4f*)(tgt + (size_t)b * (NPTS * 2));
    const v4f* ga = (const v4f*)(act + (size_t)b * (NPTS * 2));

    // ---- Kick off async copy of the target tile into LDS (ASYNCcnt path).
    // Each lane copies 4 x 16B; the DMA overlaps with all of phase 1.
    {
        uint32_t lds_base = (uint32_t)(uintptr_t)(&s_tgt[0]);
        uint64_t g_base   = (uint64_t)(uintptr_t)gt;
#pragma unroll
        for (int k = 0; k < VEC4_PER_THREAD; ++k) {
            int      idx   = tid + k * BLOCK;        // 16-byte chunk index
            uint32_t laddr = lds_base + (uint32_t)idx * 16u;
            uint64_t gaddr = g_base + (uint64_t)idx * 16u;
            asm volatile("global_load_async_to_lds_b128 %0, %1, off"
                         :: "v"(laddr), "v"(gaddr) : "memory");
        }
    }

    if (tid < 10) s_acc[tid] = 0.0f;

    // ---- Phase 1: power moments of actual_y and moments of actual_x * y^k.
    // Non-temporal: each actual point is read exactly once.
    float m1 = 0.f, m2 = 0.f, m3 = 0.f, m4 = 0.f, m5 = 0.f, m6 = 0.f;
    float X0 = 0.f, X1 = 0.f, X2 = 0.f, X3 = 0.f;
#pragma unroll
    for (int k = 0; k < VEC4_PER_THREAD; ++k) {
        v4f q = __builtin_nontemporal_load(&ga[tid + k * BLOCK]);
        {
            float x = q.x, y = q.y;
            float y2 = y * y, y3 = y2 * y;
            m1 += y; m2 += y2; m3 += y3;
            m4 = fmaf(y2, y2, m4); m5 = fmaf(y2, y3, m5); m6 = fmaf(y3, y3, m6);
            X3 += x; X2 = fmaf(x, y, X2); X1 = fmaf(x, y2, X1); X0 = fmaf(x, y3, X0);
        }
        {
            float x = q.z, y = q.w;
            float y2 = y * y, y3 = y2 * y;
            m1 += y; m2 += y2; m3 += y3;
            m4 = fmaf(y2, y2, m4); m5 = fmaf(y2, y3, m5); m6 = fmaf(y3, y3, m6);
            X3 += x; X2 = fmaf(x, y, X2); X1 = fmaf(x, y2, X1); X0 = fmaf(x, y3, X0);
        }
    }

    // ---- wave32 tree reduction of the 10 partials.
#pragma unroll
    for (int off = 16; off > 0; off >>= 1) {
        m1 += __shfl_xor(m1, off, 32);
        m2 += __shfl_xor(m2, off, 32);
        m3 += __shfl_xor(m3, off, 32);
        m4 += __shfl_xor(m4, off, 32);
        m5 += __shfl_xor(m5, off, 32);
        m6 += __shfl_xor(m6, off, 32);
        X0 += __shfl_xor(X0, off, 32);
        X1 += __shfl_xor(X1, off, 32);
        X2 += __shfl_xor(X2, off, 32);
        X3 += __shfl_xor(X3, off, 32);
    }

    __syncthreads();                       // s_acc zero-init visible
    if ((tid & 31) == 0) {                 // lane 0 of each of 8 waves
        atomicAdd(&s_acc[0], m1); atomicAdd(&s_acc[1], m2);
        atomicAdd(&s_acc[2], m3); atomicAdd(&s_acc[3], m4);
        atomicAdd(&s_acc[4], m5); atomicAdd(&s_acc[5], m6);
        atomicAdd(&s_acc[6], X0); atomicAdd(&s_acc[7], X1);
        atomicAdd(&s_acc[8], X2); atomicAdd(&s_acc[9], X3);
    }
    __syncthreads();

    // ---- SPD 4x4 solve via branch-free f64 Cholesky on one lane.
    // Z_ij = m_{6-i-j} (power-moment Gram), X_i = sum x*y^(3-i).
    if (tid == 0) {
        double m[7];
        m[0] = (double)NPTS;
        for (int i = 0; i < 6; ++i) m[i + 1] = (double)s_acc[i];

        double L[4][4], dinv[4];
#pragma unroll
        for (int i = 0; i < 4; ++i) {
#pragma unroll
            for (int j = 0; j <= i; ++j) {
                double s = m[6 - i - j];
#pragma unroll
                for (int k = 0; k < 4; ++k)
                    if (k < j) s -= L[i][k] * L[j][k];
                if (i == j) { L[i][i] = sqrt(s); dinv[i] = 1.0 / L[i][i]; }
                else        { L[i][j] = s * dinv[j]; }
            }
        }
        double w[4];
#pragma unroll
        for (int i = 0; i < 4; ++i) {
            double s = (double)s_acc[6 + i];
#pragma unroll
            for (int k = 0; k < 4; ++k)
                if (k < i) s -= L[i][k] * w[k];
            w[i] = s * dinv[i];
        }
        double beta[4];
#pragma unroll
        for (int i = 3; i >= 0; --i) {
            double s = w[i];
#pragma unroll
            for (int k = 3; k >= 0; --k)
                if (k > i) s -= L[k][i] * beta[k];
            beta[i] = s * dinv[i];
        }
#pragma unroll
        for (int i = 0; i < 4; ++i) s_beta[i] = (float)beta[i];
    }

    // ---- Wait for our async LDS writes, then barrier so every wave's tile
    // portion (and s_beta) is visible workgroup-wide.
    asm volatile("s_wait_asynccnt 0x0" ::: "memory");
    __syncthreads();

    const float b0 = s_beta[0], b1 = s_beta[1], b2 = s_beta[2], b3 = s_beta[3];

    // ---- Phase 2: pred = ((b0*y + b1)*y + b2)*y + b3 ; loss = |pred - x|.
    // Non-temporal stores: the output is written once and never re-read.
    v2f* o = (v2f*)(out + (size_t)b * NPTS);
#pragma unroll
    for (int k = 0; k < VEC4_PER_THREAD; ++k) {
        int j = tid + k * BLOCK;
        v4f q = s_tgt[j];                  // (x0,y0,x1,y1)
        float p0 = fmaf(fmaf(fmaf(b0, q.y, b1), q.y, b2), q.y, b3);
        float p1 = fmaf(fmaf(fmaf(b0, q.w, b1), q.w, b2), q.w, b3);
        v2f r;
        r.x = fabsf(p0 - q.x);
        r.y = fabsf(p1 - q.z);
        __builtin_nontemporal_store(r, &o[j]);
    }
}

extern "C" void kernel_launch(void* const* d_in, const int* in_sizes, int n_in,
                              void* d_out, int out_size, void* d_ws, size_t ws_size,
                              hipStream_t stream) {
    (void)in_sizes; (void)n_in; (void)d_ws; (void)ws_size;
    const float* tgt = (const float*)d_in[0];   // target_points (B,P,2) f32
    const float* act = (const float*)d_in[1];   // actual_points (B,P,2) f32
    float*       out = (float*)d_out;           // (B,P) f32
    const int B = out_size / NPTS;              // 4096
    lsq_loss_kernel<<<B, BLOCK, 0, stream>>>(tgt, act, out);
}